// U3Layer_18726057410938
// MI455X (gfx1250) — compile-verified
//
#include <hip/hip_runtime.h>

// ---------------------------------------------------------------------------
// U3 12-qubit circuit on 4096 states of dim 4096, fully fused:
//   - 12 single-qubit gates commute -> group 4 adjacent qubits into one
//     16x16 complex matrix (kron of four 2x2 U3s) -> 3 matmuls total.
//   - One block per state vector; state lives in LDS (padded) for all 3
//     transforms -> single HBM pass: 256 MB total @ 23.3 TB/s ~= 11 us floor.
//   - fp32 matrix math via V_WMMA_F32_16X16X4_F32 (K split into 4 chunks),
//     complex = 4 real WMMAs per chunk (A_im pre-negated; F32 WMMA has no
//     A/B NEG modifier per ISA).
// ---------------------------------------------------------------------------

typedef __attribute__((ext_vector_type(2))) float v2f;
typedef __attribute__((ext_vector_type(8))) float v8f;

#define DIM 4096
#define BATCH 4096

// LDS pad: physical = i + i/16  -> strides 1 / 16 / 256 all map to
// conflict-free bank patterns for the 16-lane gather groups.
__device__ __forceinline__ int pad(int i) { return i + (i >> 4); }

// ---------------------------------------------------------------------------
// Build the three 16x16 complex group matrices G_g = kron(U_{4g..4g+3}).
// G[g][i][j] = prod_k U_{4g+k}[bit_{3-k}(i)][bit_{3-k}(j)]
// ---------------------------------------------------------------------------
__global__ void build_gates_kernel(const float* __restrict__ thetas,
                                   float* __restrict__ gre,
                                   float* __restrict__ gim) {
  int e = threadIdx.x;            // 256 entries per group
  int i = e >> 4, j = e & 15;
  for (int g = 0; g < 3; ++g) {
    float re = 1.f, im = 0.f;
    #pragma unroll
    for (int k = 0; k < 4; ++k) {
      int q = 4 * g + k;
      float th = thetas[q * 3 + 0];
      float ph = thetas[q * 3 + 1];
      float lam = thetas[q * 3 + 2];
      float ct = cosf(th * 0.5f), st = sinf(th * 0.5f);
      int bi = (i >> (3 - k)) & 1, bj = (j >> (3 - k)) & 1;
      float ure, uim;
      if (bi == 0 && bj == 0)      { ure = ct;                 uim = 0.f; }
      else if (bi == 0 && bj == 1) { ure = -cosf(lam) * st;    uim = -sinf(lam) * st; }
      else if (bi == 1 && bj == 0) { ure =  cosf(ph) * st;     uim =  sinf(ph) * st; }
      else { float a = ph + lam;     ure =  cosf(a) * ct;       uim =  sinf(a) * ct; }
      float nre = re * ure - im * uim;
      float nim = re * uim + im * ure;
      re = nre; im = nim;
    }
    gre[g * 256 + e] = re;
    gim[g * 256 + e] = im;
  }
}

// ---------------------------------------------------------------------------
// Fused circuit kernel: one block (256 threads = 8 wave32) per state vector.
// ---------------------------------------------------------------------------
__global__ __launch_bounds__(256) void u3_fused_kernel(
    const float* __restrict__ in_re, const float* __restrict__ in_im,
    const float* __restrict__ gre, const float* __restrict__ gim,
    float* __restrict__ out) {
  __shared__ float s_re[4352];   // 4096 + pad
  __shared__ float s_im[4352];
  __shared__ float s_gre[768];   // 3 x 16 x 16 gate (re)
  __shared__ float s_gim[768];   // 3 x 16 x 16 gate (im)

  const int tid = threadIdx.x;
  const long base = (long)blockIdx.x * DIM;

  // Stage gates into LDS.
  for (int k = tid; k < 768; k += 256) { s_gre[k] = gre[k]; s_gim[k] = gim[k]; }

  // Load state (float4, coalesced) into padded LDS.
  #pragma unroll
  for (int k = 0; k < 4; ++k) {
    int i0 = k * 1024 + tid * 4;                  // i0 % 4 == 0 -> same pad block
    float4 r = *(const float4*)(in_re + base + i0);
    float4 m = *(const float4*)(in_im + base + i0);
    int p = pad(i0);
    s_re[p + 0] = r.x; s_re[p + 1] = r.y; s_re[p + 2] = r.z; s_re[p + 3] = r.w;
    s_im[p + 0] = m.x; s_im[p + 1] = m.y; s_im[p + 2] = m.z; s_im[p + 3] = m.w;
  }
  __syncthreads();

  const int lane = tid & 31;
  const int wave = tid >> 5;
  const int hi   = (lane >> 4) & 1;   // 0: lanes 0-15, 1: lanes 16-31
  const int ln   = lane & 15;

  for (int g = 0; g < 3; ++g) {
    const int S = 1 << (4 * (2 - g));            // j-stride: 256, 16, 1

    // A-matrix (gate) registers for this group, shared by both tiles.
    // 16x4 f32 A layout: lanes 0-15 -> M=ln, K={4t,4t+1}; lanes 16-31 -> K={4t+2,4t+3}.
    v2f a_re[4], a_im[4], a_imn[4];
    #pragma unroll
    for (int t = 0; t < 4; ++t) {
      int kb = 4 * t + 2 * hi;
      float r0 = s_gre[g * 256 + ln * 16 + kb];
      float r1 = s_gre[g * 256 + ln * 16 + kb + 1];
      float i0 = s_gim[g * 256 + ln * 16 + kb];
      float i1 = s_gim[g * 256 + ln * 16 + kb + 1];
      a_re[t]  = v2f{ r0,  r1};
      a_im[t]  = v2f{ i0,  i1};
      a_imn[t] = v2f{-i0, -i1};
    }

    #pragma unroll
    for (int tt = 0; tt < 2; ++tt) {
      const int tile = wave * 2 + tt;            // 8 waves x 2 = 16 tiles = 256 cols
      const int c = tile * 16 + ln;              // this lane's column (0..255)
      // element index of (column c, row j) is cbase + j*S
      const int cbase = (c / S) * (S * 16) + (c % S);

      v8f d_re = {}; v8f d_im = {};
      #pragma unroll
      for (int t = 0; t < 4; ++t) {
        int k0 = 4 * t + 2 * hi;                 // B 4x16: lanes0-15 rows {4t,4t+1}, hi {4t+2,4t+3}
        int p0 = pad(cbase + k0 * S);
        int p1 = pad(cbase + (k0 + 1) * S);
        v2f b_re = v2f{s_re[p0], s_re[p1]};
        v2f b_im = v2f{s_im[p0], s_im[p1]};
        // D_re += A_re*B_re - A_im*B_im ; D_im += A_re*B_im + A_im*B_re
        d_re = __builtin_amdgcn_wmma_f32_16x16x4_f32(false, a_re[t],  false, b_re, (short)0, d_re, false, false);
        d_re = __builtin_amdgcn_wmma_f32_16x16x4_f32(false, a_imn[t], false, b_im, (short)0, d_re, false, false);
        d_im = __builtin_amdgcn_wmma_f32_16x16x4_f32(false, a_re[t],  false, b_im, (short)0, d_im, false, false);
        d_im = __builtin_amdgcn_wmma_f32_16x16x4_f32(false, a_im[t],  false, b_re, (short)0, d_im, false, false);
      }
      // D 16x16: VGPR v -> row (v + 8*hi), col = ln. Tiles own disjoint
      // columns, so no intra-group barrier is needed.
      #pragma unroll
      for (int v = 0; v < 8; ++v) {
        int m = v + 8 * hi;
        int p = pad(cbase + m * S);
        s_re[p] = d_re[v];
        s_im[p] = d_im[v];
      }
    }
    __syncthreads();                             // columns regroup between passes
  }

  // Store interleaved (re, im): out[(base+i)*2 + {0,1}], float4 = 2 elements.
  #pragma unroll
  for (int k = 0; k < 8; ++k) {
    int i0 = k * 512 + tid * 2;                  // even -> same pad block for i0, i0+1
    int p = pad(i0);
    float4 v;
    v.x = s_re[p];     v.y = s_im[p];
    v.z = s_re[p + 1]; v.w = s_im[p + 1];
    *(float4*)(out + (base + i0) * 2) = v;
  }
}

// ---------------------------------------------------------------------------
extern "C" void kernel_launch(void* const* d_in, const int* in_sizes, int n_in,
                              void* d_out, int out_size, void* d_ws, size_t ws_size,
                              hipStream_t stream) {
  const float* thetas = (const float*)d_in[0];   // (12, 3)
  const float* s_re   = (const float*)d_in[1];   // (4096, 4096, 1)
  const float* s_im   = (const float*)d_in[2];   // (4096, 4096, 1)
  float* out = (float*)d_out;                    // (4096, 4096, 2)

  float* gre = (float*)d_ws;                     // 3*256 floats
  float* gim = gre + 768;

  build_gates_kernel<<<1, 256, 0, stream>>>(thetas, gre, gim);
  u3_fused_kernel<<<BATCH, 256, 0, stream>>>(s_re, s_im, gre, gim, out);
}